// lta_16045997818333
// MI455X (gfx1250) — compile-verified
//
#include <hip/hip_runtime.h>

// EMA scan  lta[t] = lta[t-1] + w*(x[t]-lta[t-1])  over wave[256][60000][3] f32.
// Blocked linear-recurrence formulation: per 16-step block,
//   Y(16x16) = M(16x16, lower-tri) * X(16 time x 16 seq) + pow_vec (x) carry
// executed as 4 chained V_WMMA_F32_16X16X4_F32 (full f32 precision).
// Time is chunked; each chunk re-establishes its carry with a 1024-step warmup
// (residual (1-w)^1024 ~ 2e-7 relative -> below f32 scan rounding noise).
// All global accesses use running base pointers + immediate offsets so the
// hot loop is loads/WMMA/stores with minimal address VALU.

#define LTA_W    0.015f
#define OMW      (1.0f - LTA_W)

#define BATCH    256
#define TLEN     60000
#define CHN      3
#define NSEQ     (BATCH * CHN)        // 768 independent sequences
#define NTILE    (NSEQ / 16)          // 48 tiles of 16 sequences
#define NCHUNK   50
#define LCHUNK   (TLEN / NCHUNK)      // 1200 steps per chunk (75 blocks of 16)
#define WARM     1024                 // warmup steps (64 blocks)
#define PREF     768                  // prefetch distance in timesteps
#define BLKF     (16 * CHN)           // floats advanced per 16-step block

typedef float v2f __attribute__((ext_vector_type(2)));
typedef float v8f __attribute__((ext_vector_type(8)));

__global__ __launch_bounds__(32)
void lta_ema_wmma_kernel(const float* __restrict__ wave, float* __restrict__ out) {
    const int lane = threadIdx.x;           // one wave32 per workgroup
    const bool hi  = lane >= 16;
    const int  col = lane & 15;             // N column for B/C/D; M row for A

    // ---- powers of (1-w) ----
    float powv[17];
    powv[0] = 1.0f;
#pragma unroll
    for (int k = 1; k <= 16; ++k) powv[k] = powv[k - 1] * OMW;

    float pl[8];                            // pl[r] = (1-w)^(r+1)
    pl[0] = OMW;
#pragma unroll
    for (int r = 1; r < 8; ++r) pl[r] = pl[r - 1] * OMW;
    const float hic = hi ? (pl[7] * OMW) : 1.0f;   // (1-w)^9 for the high rows

    // ---- A operand: M[i][j] = (j<=i) ? w*(1-w)^(i-j) : 0, 16x4 per WMMA ----
    // lane&15 = row; VGPR .x/.y = K pair; lane half holds K{0,1} vs K{2,3}.
    v2f Amat[4];
#pragma unroll
    for (int kk = 0; kk < 4; ++kk) {
        const int ks = kk * 4 + (hi ? 2 : 0);
        Amat[kk].x = (ks     <= col) ? (LTA_W * powv[col - ks])     : 0.0f;
        Amat[kk].y = (ks + 1 <= col) ? (LTA_W * powv[col - ks - 1]) : 0.0f;
    }

    // ---- which 16 sequences / which time chunk this wave owns ----
    const int seq = blockIdx.x * 16 + col;        // (b, c) pair
    const int b   = seq / CHN;
    const int c   = seq % CHN;
    const size_t seqbase = (size_t)b * ((size_t)TLEN * CHN) + (size_t)c;

    const int t0     = blockIdx.y * LCHUNK;
    const int tend   = t0 + LCHUNK;
    const int tstart = (blockIdx.y == 0) ? 0 : (t0 - WARM);

    // Running pointers: lane-half K/row offsets folded into the base so the
    // per-block accesses are compile-time immediate offsets.
    const float* __restrict__ pb =
        wave + seqbase + (size_t)tstart * CHN + (hi ? 2 : 0) * CHN;
    float* __restrict__ ps =
        out + seqbase + (size_t)t0 * CHN + (hi ? 8 : 0) * CHN;

    float carry = 0.0f;

    // One 16-step block: 4 chained f32 WMMAs + carry broadcast.
    auto block16 = [&](bool emit, int tb) {
        const float gc = carry * hic;       // seed C[i][n] = (1-w)^(i+1)*c[n]; rows 8..15 get (1-w)^(i+2)
        v8f acc;
#pragma unroll
        for (int r = 0; r < 8; ++r) acc[r] = pl[r] * gc;

#pragma unroll
        for (int kk = 0; kk < 4; ++kk) {
            v2f bv;
            bv.x = pb[kk * 4 * CHN];        // byte offsets 0,48,96,144
            bv.y = pb[kk * 4 * CHN + CHN];  // byte offsets 12,60,108,156
            acc = __builtin_amdgcn_wmma_f32_16x16x4_f32(
                false, Amat[kk], false, bv, (short)0, acc, false, false);
        }

        if (tb + PREF + 1 < TLEN)           // stream-ahead prefetch (imm offset)
            __builtin_prefetch(pb + PREF * CHN, 0, 3);

        if (emit) {
#pragma unroll
            for (int r = 0; r < 8; ++r)
                ps[r * CHN] = acc[r];       // byte offsets 0..84
            ps += BLKF;
        }
        pb += BLKF;

        // New carry = Y row 15 = D[7] of lanes 16..31; broadcast per column
        // via lane-permute (no LDS memory touched, EXEC all ones).
        carry = __int_as_float(
            __builtin_amdgcn_ds_bpermute((16 + col) << 2,
                                         __float_as_int(acc[7])));
    };

    // Warmup loop (carry-only, no stores), then emit loop.
    for (int tb = tstart; tb < t0; tb += 16) block16(false, tb);
    for (int tb = t0;     tb < tend; tb += 16) block16(true, tb);
}

extern "C" void kernel_launch(void* const* d_in, const int* in_sizes, int n_in,
                              void* d_out, int out_size, void* d_ws, size_t ws_size,
                              hipStream_t stream) {
    (void)in_sizes; (void)n_in; (void)out_size; (void)d_ws; (void)ws_size;
    const float* wave = (const float*)d_in[0];
    float*       out  = (float*)d_out;

    dim3 grid(NTILE, NCHUNK);   // 48 sequence-tiles x 50 time-chunks = 2400 waves
    dim3 block(32);             // exactly one wave32 (WMMA needs full EXEC)
    lta_ema_wmma_kernel<<<grid, block, 0, stream>>>(wave, out);
}